// GATLayer_36223754174836
// MI455X (gfx1250) — compile-verified
//
#include <hip/hip_runtime.h>
#include <hip/hip_bf16.h>

#define NN 4096
#define CH 256
#define NH 8
#define HD 32

typedef float v2f __attribute__((ext_vector_type(2)));
typedef float v8f __attribute__((ext_vector_type(8)));

static __device__ __forceinline__ v8f wmma_f32x4(v2f a, v2f b, v8f c) {
    // D = A(16x4 f32) * B(4x16 f32) + C(16x16 f32)
    return __builtin_amdgcn_wmma_f32_16x16x4_f32(
        /*neg_a=*/false, a, /*neg_b=*/false, b,
        /*c_mod=*/(short)0, c, /*reuse_a=*/false, /*reuse_b=*/false);
}

// ---------------------------------------------------------------------------
// Kernel 1: Q/K/V projection.  One wave32 computes one 16x16 output tile of
// (X @ W + b) via 64 chained V_WMMA_F32_16X16X4_F32 ops.
// grid = (N/16, CH/16, 3), block = 32.
// ---------------------------------------------------------------------------
__global__ __launch_bounds__(32) void qkv_kernel(
    const float* __restrict__ X,
    const float* __restrict__ Wq, const float* __restrict__ bq,
    const float* __restrict__ Wk, const float* __restrict__ bk,
    const float* __restrict__ Wv, const float* __restrict__ bv,
    float* __restrict__ qkv /* ws: [3][N][CH] */) {
    const int i0   = blockIdx.x * 16;
    const int col0 = blockIdx.y * 16;
    const int mat  = blockIdx.z;
    const float* W = (mat == 0) ? Wq : (mat == 1) ? Wk : Wv;
    const float* b = (mat == 0) ? bq : (mat == 1) ? bk : bv;
    float* outp = qkv + (size_t)mat * NN * CH;

    const int lane = threadIdx.x & 31;
    const int m    = lane & 15;   // row (A) / col (B/D) within tile
    const int half = lane >> 4;   // K-half selector

    v8f c = {0.f, 0.f, 0.f, 0.f, 0.f, 0.f, 0.f, 0.f};
    for (int kk = 0; kk < CH; kk += 4) {
        v2f a, bb;
        const float* xr = X + (size_t)(i0 + m) * CH + kk + 2 * half;
        a.x  = xr[0];
        a.y  = xr[1];
        bb.x = W[(size_t)(kk + 2 * half)     * CH + col0 + m];
        bb.y = W[(size_t)(kk + 2 * half + 1) * CH + col0 + m];
        c = wmma_f32x4(a, bb, c);
    }
    const float bias = b[col0 + m];
#pragma unroll
    for (int r = 0; r < 8; ++r)
        outp[(size_t)(i0 + r + 8 * half) * CH + col0 + m] = c[r] + bias;
}

// ---------------------------------------------------------------------------
// Kernel 2: fused attention.  Block = 256 threads = 8 waves; wave w == head h.
// Block owns 16 rows (i-tile), loops over all j in 16-wide chunks:
//   scores (WMMA) -> LDS -> +rel, softmax over heads -> attn out + LDS ->
//   AV accumulation (WMMA, persistent accumulators) -> head-reduce -> MLP.
// grid = (N/16), block = 256.
// ---------------------------------------------------------------------------
__global__ __launch_bounds__(256) void attn_kernel(
    const float* __restrict__ q, const float* __restrict__ k,
    const float* __restrict__ v, const float* __restrict__ rel,
    const float* __restrict__ W1, const float* __restrict__ b1,
    const float* __restrict__ W2, const float* __restrict__ b2,
    float* __restrict__ out,        // [N, CH]
    float* __restrict__ attn_g) {   // [N, N, NH]
    __shared__ float lds_s[NH * 16 * 16];  // scores/attn tile; later attn_out[16][32]
    __shared__ float lds_p[16 * 256];      // head partials [8][16][32]; later hidden[16][256]

    const int i0   = blockIdx.x * 16;
    const int tid  = threadIdx.x;
    const int h    = tid >> 5;     // wave id == head
    const int lane = tid & 31;
    const int m    = lane & 15;
    const int half = lane >> 4;

    // Q A-fragments for this head, resident for the whole j loop.
    // q_reshaped[i, c, h] == q_flat[i, c*8 + h],  c in [0,32)
    v2f aq[8];
#pragma unroll
    for (int s = 0; s < 8; ++s) {
        const int c = 4 * s + 2 * half;
        aq[s].x = q[(size_t)(i0 + m) * CH + c * 8 + h];
        aq[s].y = q[(size_t)(i0 + m) * CH + (c + 1) * 8 + h];
    }

    v8f acc_lo = {0.f, 0.f, 0.f, 0.f, 0.f, 0.f, 0.f, 0.f};
    v8f acc_hi = {0.f, 0.f, 0.f, 0.f, 0.f, 0.f, 0.f, 0.f};

    const int il = tid >> 4;   // softmax pair: local row
    const int jl = tid & 15;   // softmax pair: local col

    for (int j0 = 0; j0 < NN; j0 += 16) {
        // ---- scores tile for head h: Q_h[16x32] @ K_h^T[32x16] ----
        v8f sc = {0.f, 0.f, 0.f, 0.f, 0.f, 0.f, 0.f, 0.f};
#pragma unroll
        for (int s = 0; s < 8; ++s) {
            const int c = 4 * s + 2 * half;
            v2f bk_;
            bk_.x = k[(size_t)(j0 + m) * CH + c * 8 + h];
            bk_.y = k[(size_t)(j0 + m) * CH + (c + 1) * 8 + h];
            sc = wmma_f32x4(aq[s], bk_, sc);
        }
#pragma unroll
        for (int r = 0; r < 8; ++r)
            lds_s[h * 256 + (r + 8 * half) * 16 + m] = sc[r];
        __syncthreads();

        // ---- + rel, softmax over the 8 heads for pair (il, jl) ----
        {
            const size_t ro = ((size_t)(i0 + il) * NN + (size_t)(j0 + jl)) * NH;
            const float4* relp = (const float4*)(rel + ro);
            __builtin_prefetch(rel + ro + 16 * NH, 0, 0);  // next j-chunk
            const float4 r0 = relp[0];
            const float4 r1 = relp[1];
            float sv[8];
            sv[0] = lds_s[0 * 256 + il * 16 + jl] + r0.x;
            sv[1] = lds_s[1 * 256 + il * 16 + jl] + r0.y;
            sv[2] = lds_s[2 * 256 + il * 16 + jl] + r0.z;
            sv[3] = lds_s[3 * 256 + il * 16 + jl] + r0.w;
            sv[4] = lds_s[4 * 256 + il * 16 + jl] + r1.x;
            sv[5] = lds_s[5 * 256 + il * 16 + jl] + r1.y;
            sv[6] = lds_s[6 * 256 + il * 16 + jl] + r1.z;
            sv[7] = lds_s[7 * 256 + il * 16 + jl] + r1.w;
            float mx = sv[0];
#pragma unroll
            for (int hh = 1; hh < 8; ++hh) mx = fmaxf(mx, sv[hh]);
            float sum = 0.f;
#pragma unroll
            for (int hh = 0; hh < 8; ++hh) { sv[hh] = __expf(sv[hh] - mx); sum += sv[hh]; }
            const float inv = __frcp_rn(sum);
#pragma unroll
            for (int hh = 0; hh < 8; ++hh) {
                sv[hh] *= inv;
                lds_s[hh * 256 + il * 16 + jl] = sv[hh];
            }
            float4* ap = (float4*)(attn_g + ro);
            ap[0] = make_float4(sv[0], sv[1], sv[2], sv[3]);
            ap[1] = make_float4(sv[4], sv[5], sv[6], sv[7]);
        }
        __syncthreads();

        // ---- AV: attn_h[16x16] @ V'_h[16x32],  V'[j,h,d] = v_flat[j, d*8+h] ----
#pragma unroll
        for (int s = 0; s < 4; ++s) {
            const int kk = 4 * s + 2 * half;
            v2f aa, bl, bh;
            aa.x = lds_s[h * 256 + m * 16 + kk];
            aa.y = lds_s[h * 256 + m * 16 + kk + 1];
            bl.x = v[(size_t)(j0 + kk)     * CH + m * 8 + h];
            bl.y = v[(size_t)(j0 + kk + 1) * CH + m * 8 + h];
            bh.x = v[(size_t)(j0 + kk)     * CH + (m + 16) * 8 + h];
            bh.y = v[(size_t)(j0 + kk + 1) * CH + (m + 16) * 8 + h];
            acc_lo = wmma_f32x4(aa, bl, acc_lo);
            acc_hi = wmma_f32x4(aa, bh, acc_hi);
        }
        __syncthreads();  // lds_s reused next iteration
    }

    // ---- reduce head partials: attn_out[16][32] ----
#pragma unroll
    for (int r = 0; r < 8; ++r) {
        const int M = r + 8 * half;
        lds_p[h * 512 + M * 32 + m]      = acc_lo[r];
        lds_p[h * 512 + M * 32 + 16 + m] = acc_hi[r];
    }
    __syncthreads();
    for (int e = tid; e < 512; e += 256) {
        float s = 0.f;
#pragma unroll
        for (int hh = 0; hh < 8; ++hh) s += lds_p[hh * 512 + e];
        lds_s[e] = s;   // attn_out flattened [16][32]
    }
    __syncthreads();

    // ---- MLP: hidden = relu(attn_out @ W1 + b1); out = hidden @ W2 + b2 ----
    float hid[16];
    const int o = tid;  // output column
#pragma unroll 4
    for (int mm = 0; mm < 16; ++mm) {
        float a = b1[o];
        for (int c = 0; c < HD; ++c)
            a = fmaf(lds_s[mm * 32 + c], W1[(size_t)c * CH + o], a);
        hid[mm] = fmaxf(a, 0.f);
    }
#pragma unroll
    for (int mm = 0; mm < 16; ++mm) lds_p[mm * 256 + o] = hid[mm];
    __syncthreads();
    for (int mm = 0; mm < 16; ++mm) {
        float a = b2[o];
        for (int c = 0; c < 256; ++c)
            a = fmaf(lds_p[mm * 256 + c], W2[(size_t)c * CH + o], a);
        out[(size_t)(i0 + mm) * CH + o] = a;
    }
}

// ---------------------------------------------------------------------------
extern "C" void kernel_launch(void* const* d_in, const int* in_sizes, int n_in,
                              void* d_out, int out_size, void* d_ws, size_t ws_size,
                              hipStream_t stream) {
    const float* X   = (const float*)d_in[0];
    const float* rel = (const float*)d_in[1];
    // d_in[2] = num_nodes (compile-time NN here)
    const float* Wq = (const float*)d_in[3];
    const float* bq = (const float*)d_in[4];
    const float* Wk = (const float*)d_in[5];
    const float* bk = (const float*)d_in[6];
    const float* Wv = (const float*)d_in[7];
    const float* bv = (const float*)d_in[8];
    const float* W1 = (const float*)d_in[9];
    const float* b1 = (const float*)d_in[10];
    const float* W2 = (const float*)d_in[11];
    const float* b2 = (const float*)d_in[12];

    float* qkv  = (float*)d_ws;                       // [3][N][CH] = 12 MB
    float* outp = (float*)d_out;                      // [N][CH]
    float* attn = outp + (size_t)NN * CH;             // [N][N][NH]

    qkv_kernel<<<dim3(NN / 16, CH / 16, 3), 32, 0, stream>>>(
        X, Wq, bq, Wk, bk, Wv, bv, qkv);

    attn_kernel<<<dim3(NN / 16), 256, 0, stream>>>(
        qkv, qkv + (size_t)NN * CH, qkv + 2 * (size_t)NN * CH, rel,
        W1, b1, W2, b2, outp, attn);
}